// Model_10264971837758
// MI455X (gfx1250) — compile-verified
//
#include <hip/hip_runtime.h>

#define N_NODES 100000
#define N_EDGES 1600000
#define D 64
#define MPNN_STEPS 4

typedef float v2f __attribute__((ext_vector_type(2)));
typedef float v8f __attribute__((ext_vector_type(8)));

// ---------------------------------------------------------------------------
// init: zero msg (N_NODES*D), deg (N_NODES), gsum (D)
// ---------------------------------------------------------------------------
__global__ void __launch_bounds__(256) init_kernel(float* __restrict__ msg,
                                                   float* __restrict__ deg,
                                                   float* __restrict__ gsum) {
  int total = N_NODES * D;
  for (int i = blockIdx.x * blockDim.x + threadIdx.x; i < total;
       i += gridDim.x * blockDim.x) {
    msg[i] = 0.0f;
    if (i < N_NODES) deg[i] = 0.0f;
    if (i < D) gsum[i] = 0.0f;
  }
}

// ---------------------------------------------------------------------------
// degree: atomic add 1 at dst
// ---------------------------------------------------------------------------
__global__ void __launch_bounds__(256) deg_kernel(const int* __restrict__ dst,
                                                  float* __restrict__ deg) {
  int e = blockIdx.x * blockDim.x + threadIdx.x;
  if (e < N_EDGES) atomicAdd(&deg[dst[e]], 1.0f);
}

__global__ void __launch_bounds__(256) inv_kernel(const float* __restrict__ deg,
                                                  float* __restrict__ inv) {
  int i = blockIdx.x * blockDim.x + threadIdx.x;
  if (i < N_NODES) inv[i] = 1.0f / fmaxf(deg[i], 1.0f);
}

// ---------------------------------------------------------------------------
// gemm_in: H = relu(X @ W + b),  X:[N,64], W:[64,64], one 16x16 tile per wave
// using V_WMMA_F32_16X16X4_F32, K-loop of 16 (64/4).
// A(16x4 f32): lane L<16 -> row L, k = {0,1}; lane L+16 -> row L, k = {2,3}
// B(4x16 f32): lane L<16 -> col L, k = {0,1}; lane L+16 -> col L, k = {2,3}
// C/D(16x16 f32): vgpr v, lane<16 -> (M=v, N=lane); lane>=16 -> (M=v+8, N=lane-16)
// ---------------------------------------------------------------------------
__global__ void __launch_bounds__(256) gemm_in_kernel(const float* __restrict__ X,
                                                      const float* __restrict__ W,
                                                      const float* __restrict__ bias,
                                                      float* __restrict__ H) {
  int wave = (blockIdx.x * blockDim.x + threadIdx.x) >> 5;
  int lane = threadIdx.x & 31;
  int tileRow = wave >> 2;   // 6250 row tiles
  int tileCol = wave & 3;    // 4 col tiles
  if (tileRow * 16 >= N_NODES) return;  // wave-uniform; EXEC stays all-ones

  int r = lane & 15;
  int khalf = lane >> 4;                 // 0 or 1
  const float* arow = X + (size_t)(tileRow * 16 + r) * D;
  int bcol = tileCol * 16 + r;

  v8f c = {};
#pragma unroll
  for (int kc = 0; kc < 16; ++kc) {
    int k0 = kc * 4 + khalf * 2;
    v2f a, b;
    a.x = arow[k0];
    a.y = arow[k0 + 1];
    b.x = W[(size_t)k0 * D + bcol];
    b.y = W[(size_t)(k0 + 1) * D + bcol];
    c = __builtin_amdgcn_wmma_f32_16x16x4_f32(false, a, false, b,
                                              (short)0, c, false, false);
  }

  int outcol = tileCol * 16 + (lane & 15);
  int rbase = tileRow * 16 + (lane >> 4) * 8;
  float bv = bias[outcol];
#pragma unroll
  for (int v = 0; v < 8; ++v) {
    float val = c[v] + bv;
    H[(size_t)(rbase + v) * D + outcol] = fmaxf(val, 0.0f);
  }
}

// ---------------------------------------------------------------------------
// scatter: one wave per edge; each lane moves float2 (32 lanes * 2 = 64 dims)
// ---------------------------------------------------------------------------
__global__ void __launch_bounds__(256) scatter_kernel(const int* __restrict__ src,
                                                      const int* __restrict__ dst,
                                                      const float* __restrict__ h,
                                                      float* __restrict__ msg) {
  int tid = blockIdx.x * blockDim.x + threadIdx.x;
  int e = tid >> 5;
  if (e >= N_EDGES) return;
  int lane = tid & 31;
  int s = src[e];
  int d = dst[e];
  const float2* hp = (const float2*)(h + (size_t)s * D);
  float2 v = hp[lane];
  float* mp = msg + (size_t)d * D + lane * 2;
  atomicAdd(&mp[0], v.x);
  atomicAdd(&mp[1], v.y);
}

// ---------------------------------------------------------------------------
// update: h = (h + msg*inv_deg) * 0.5 ; re-zero msg for the next round
// ---------------------------------------------------------------------------
__global__ void __launch_bounds__(256) update_kernel(float* __restrict__ h,
                                                     float* __restrict__ msg,
                                                     const float* __restrict__ inv) {
  int i = blockIdx.x * blockDim.x + threadIdx.x;
  if (i >= N_NODES * D) return;
  int node = i >> 6;
  float m = msg[i] * inv[node];
  h[i] = (h[i] + m) * 0.5f;
  msg[i] = 0.0f;
}

// ---------------------------------------------------------------------------
// gemm_ro + fused readout: colsum += sum_rows( relu(H) @ W_ro ), via WMMA.
// Per-lane: sum the 8 accumulator rows (same column), LDS-reduce per block,
// then 64 global atomics per block.
// ---------------------------------------------------------------------------
__global__ void __launch_bounds__(256) gemm_ro_kernel(const float* __restrict__ H,
                                                      const float* __restrict__ W,
                                                      float* __restrict__ gsum) {
  __shared__ float sgs[D];
  if (threadIdx.x < D) sgs[threadIdx.x] = 0.0f;
  __syncthreads();

  int wave = (blockIdx.x * blockDim.x + threadIdx.x) >> 5;
  int lane = threadIdx.x & 31;
  int tileRow = wave >> 2;
  int tileCol = wave & 3;

  if (tileRow * 16 < N_NODES) {
    int r = lane & 15;
    int khalf = lane >> 4;
    const float* arow = H + (size_t)(tileRow * 16 + r) * D;
    int bcol = tileCol * 16 + r;

    v8f c = {};
#pragma unroll
    for (int kc = 0; kc < 16; ++kc) {
      int k0 = kc * 4 + khalf * 2;
      v2f a, b;
      a.x = fmaxf(arow[k0], 0.0f);       // relu fused on A operand
      a.y = fmaxf(arow[k0 + 1], 0.0f);
      b.x = W[(size_t)k0 * D + bcol];
      b.y = W[(size_t)(k0 + 1) * D + bcol];
      c = __builtin_amdgcn_wmma_f32_16x16x4_f32(false, a, false, b,
                                                (short)0, c, false, false);
    }

    float colsum = 0.0f;
#pragma unroll
    for (int v = 0; v < 8; ++v) colsum += c[v];
    int outcol = tileCol * 16 + (lane & 15);
    atomicAdd(&sgs[outcol], colsum);
  }
  __syncthreads();
  if (threadIdx.x < D) atomicAdd(&gsum[threadIdx.x], sgs[threadIdx.x]);
}

// ---------------------------------------------------------------------------
// final: g = relu(colsum/N + b_ro); out = g @ W_pred + b_pred  (single block)
// ---------------------------------------------------------------------------
__global__ void __launch_bounds__(64) final_kernel(const float* __restrict__ gsum,
                                                   const float* __restrict__ b_ro,
                                                   const float* __restrict__ W_pred,
                                                   const float* __restrict__ b_pred,
                                                   float* __restrict__ out) {
  __shared__ float s[D];
  int t = threadIdx.x;
  float g = gsum[t] * (1.0f / (float)N_NODES) + b_ro[t];
  g = fmaxf(g, 0.0f);
  s[t] = g * W_pred[t];
  __syncthreads();
  if (t == 0) {
    float acc = 0.0f;
#pragma unroll
    for (int i = 0; i < D; ++i) acc += s[i];
    out[0] = acc + b_pred[0];
  }
}

// ---------------------------------------------------------------------------
extern "C" void kernel_launch(void* const* d_in, const int* in_sizes, int n_in,
                              void* d_out, int out_size, void* d_ws, size_t ws_size,
                              hipStream_t stream) {
  const float* x      = (const float*)d_in[0];
  const int*   eidx   = (const int*)d_in[1];
  const float* W_in   = (const float*)d_in[2];
  const float* b_in   = (const float*)d_in[3];
  const float* W_ro   = (const float*)d_in[4];
  const float* b_ro   = (const float*)d_in[5];
  const float* W_pred = (const float*)d_in[6];
  const float* b_pred = (const float*)d_in[7];
  float* out = (float*)d_out;

  const int* src = eidx;
  const int* dst = eidx + N_EDGES;

  // workspace layout (all 128B aligned)
  char* ws = (char*)d_ws;
  size_t off = 0;
  float* h    = (float*)(ws + off); off += (size_t)N_NODES * D * sizeof(float);
  float* msg  = (float*)(ws + off); off += (size_t)N_NODES * D * sizeof(float);
  float* deg  = (float*)(ws + off); off += (size_t)N_NODES * sizeof(float);
  float* inv  = (float*)(ws + off); off += (size_t)N_NODES * sizeof(float);
  float* gsum = (float*)(ws + off); off += 256;

  const int T = 256;
  int elems = N_NODES * D;                       // 6,400,000
  int blk_elems = (elems + T - 1) / T;           // 25,000
  int blk_edges = (N_EDGES + T - 1) / T;         // 6,250
  int blk_nodes = (N_NODES + T - 1) / T;         // 391
  int n_tiles = (N_NODES / 16) * 4;              // 25,000 waves
  int blk_gemm = n_tiles / 8;                    // 3,125 blocks (8 waves/block)
  int blk_scat = (N_EDGES * 32) / T;             // 200,000 blocks

  init_kernel<<<blk_elems, T, 0, stream>>>(msg, deg, gsum);
  deg_kernel<<<blk_edges, T, 0, stream>>>(dst, deg);
  inv_kernel<<<blk_nodes, T, 0, stream>>>(deg, inv);

  gemm_in_kernel<<<blk_gemm, T, 0, stream>>>(x, W_in, b_in, h);

  for (int step = 0; step < MPNN_STEPS; ++step) {
    scatter_kernel<<<blk_scat, T, 0, stream>>>(src, dst, h, msg);
    update_kernel<<<blk_elems, T, 0, stream>>>(h, msg, inv);
  }

  gemm_ro_kernel<<<blk_gemm, T, 0, stream>>>(h, W_ro, gsum);
  final_kernel<<<1, 64, 0, stream>>>(gsum, b_ro, W_pred, b_pred, out);
}